// TransitionDown_38173669327299
// MI455X (gfx1250) — compile-verified
//
#include <hip/hip_runtime.h>

// ---------------------------------------------------------------------------
// TransitionDown (PointNet++) for MI455X / gfx1250, wave32.
// B=16, N=4096, CX=64, COUT=128, CAUX=32, RATIO=0.25 -> M=1024, K=16
// ---------------------------------------------------------------------------

#define B_    16
#define N_    4096
#define CX_   64
#define COUT_ 128
#define CAUX_ 32
#define K_    16
#define M_    (N_ / 4)          // 1024
#define P_    (B_ * N_)         // 65536
#define BM_   (B_ * M_)         // 16384
#define EPS_  1e-5f

typedef float v2f __attribute__((ext_vector_type(2)));
typedef float v8f __attribute__((ext_vector_type(8)));

// Raw 32-bit LDS byte offset of a shared-memory pointer (for async-to-LDS asm).
typedef __attribute__((address_space(3))) char lds_char;
static __device__ __forceinline__ unsigned lds_off_u32(void* p) {
    return (unsigned)(unsigned long long)(lds_char*)p;
}

// ---------------------------------------------------------------------------
// Zero the BN stat accumulators (must be re-zeroed every launch: deterministic).
// ---------------------------------------------------------------------------
__global__ void init_stats_kernel(float* __restrict__ sum, float* __restrict__ sumsq) {
    int t = threadIdx.x;
    if (t < COUT_) { sum[t] = 0.0f; sumsq[t] = 0.0f; }
}

// ---------------------------------------------------------------------------
// Farthest-point sampling: one 1024-thread block per cloud, 4 points/thread.
// Sequential M-1 argmax steps; wave32 shfl reduce + 32-entry LDS reduce.
// Tie-break: lowest index (matches jnp.argmax).
// Also emits centers (ws), sub_pos, sub_batch, id_clusters (outputs).
// ---------------------------------------------------------------------------
__global__ __launch_bounds__(1024) void fps_kernel(
    const float* __restrict__ pos, float* __restrict__ centers,
    float* __restrict__ out_sub_pos, float* __restrict__ out_sub_batch,
    float* __restrict__ out_id)
{
    __shared__ int   lidx[M_];
    __shared__ float redD[32];
    __shared__ int   redI[32];

    const int cloud = blockIdx.x;
    const int base  = cloud * N_;
    const int tid   = threadIdx.x;
    const int wid   = tid >> 5;
    const int lane  = tid & 31;

    float px[4], py[4], pz[4], mind[4];
    const float x0 = pos[(size_t)base * 3 + 0];
    const float y0 = pos[(size_t)base * 3 + 1];
    const float z0 = pos[(size_t)base * 3 + 2];
#pragma unroll
    for (int j = 0; j < 4; ++j) {
        int p = tid + j * 1024;
        px[j] = pos[(size_t)(base + p) * 3 + 0];
        py[j] = pos[(size_t)(base + p) * 3 + 1];
        pz[j] = pos[(size_t)(base + p) * 3 + 2];
        float dx = px[j] - x0, dy = py[j] - y0, dz = pz[j] - z0;
        mind[j] = dx * dx + dy * dy + dz * dz;
    }
    if (tid == 0) lidx[0] = 0;

    for (int step = 1; step < M_; ++step) {
        // local argmax over this thread's 4 points (lowest index wins ties)
        float bd = mind[0]; int bi = tid;
#pragma unroll
        for (int j = 1; j < 4; ++j)
            if (mind[j] > bd) { bd = mind[j]; bi = tid + j * 1024; }
        // wave32 argmax reduce
#pragma unroll
        for (int off = 16; off >= 1; off >>= 1) {
            float od = __shfl_xor(bd, off, 32);
            int   oi = __shfl_xor(bi, off, 32);
            if (od > bd || (od == bd && oi < bi)) { bd = od; bi = oi; }
        }
        if (lane == 0) { redD[wid] = bd; redI[wid] = bi; }
        __syncthreads();
        if (wid == 0) {
            float bd2 = redD[lane]; int bi2 = redI[lane];
#pragma unroll
            for (int off = 16; off >= 1; off >>= 1) {
                float od = __shfl_xor(bd2, off, 32);
                int   oi = __shfl_xor(bi2, off, 32);
                if (od > bd2 || (od == bd2 && oi < bi2)) { bd2 = od; bi2 = oi; }
            }
            if (lane == 0) lidx[step] = bi2;
        }
        __syncthreads();
        const int sel = lidx[step];
        const float cx = pos[(size_t)(base + sel) * 3 + 0];
        const float cy = pos[(size_t)(base + sel) * 3 + 1];
        const float cz = pos[(size_t)(base + sel) * 3 + 2];
#pragma unroll
        for (int j = 0; j < 4; ++j) {
            float dx = px[j] - cx, dy = py[j] - cy, dz = pz[j] - cz;
            mind[j] = fminf(mind[j], dx * dx + dy * dy + dz * dz);
        }
    }
    __syncthreads();
    for (int i = tid; i < M_; i += 1024) {
        int li = lidx[i];
        int g  = base + li;
        float X = pos[(size_t)g * 3 + 0];
        float Y = pos[(size_t)g * 3 + 1];
        float Z = pos[(size_t)g * 3 + 2];
        int cl = cloud * M_ + i;
        centers[(size_t)cl * 3 + 0] = X;
        centers[(size_t)cl * 3 + 1] = Y;
        centers[(size_t)cl * 3 + 2] = Z;
        out_sub_pos[(size_t)cl * 3 + 0] = X;
        out_sub_pos[(size_t)cl * 3 + 1] = Y;
        out_sub_pos[(size_t)cl * 3 + 2] = Z;
        out_sub_batch[cl] = (float)cloud;   // int32 in ref; d_out is a float buffer
        out_id[cl]        = (float)g;       // int32 in ref; d_out is a float buffer
    }
}

// ---------------------------------------------------------------------------
// h = x @ W + b with V_WMMA_F32_16X16X4_F32, fused BN-stat accumulation.
// One block = 16 rows x 128 cols; 8 waves, one 16-col tile each; A in LDS.
// A(16x4): lane<16 rows, VGPR v holds K=2*half+v.  B(4x16): VGPR v holds
// K=2*half+v at N=lane&15.  C/D: VGPR r -> row 8*half+r, col lane&15.
// (Synchronous LDS staging kept here: the stride-65 pad is what makes the
//  per-WMMA ds_load_2addr fragment reads bank-conflict-free.)
// ---------------------------------------------------------------------------
__global__ __launch_bounds__(256) void gemm_bn_kernel(
    const float* __restrict__ x, const float* __restrict__ W,
    const float* __restrict__ bias, float* __restrict__ h,
    float* __restrict__ sum, float* __restrict__ sumsq)
{
    __shared__ float As[16 * 65];   // stride 65: conflict-free column reads
    const int row0 = blockIdx.x * 16;
    const int tid  = threadIdx.x;
    const int wave = tid >> 5;
    const int lane = tid & 31;
    const int half = lane >> 4;
    const int lm   = lane & 15;
    const int col0 = wave * 16;

    for (int e = tid; e < 16 * CX_; e += 256) {
        int r = e >> 6, c = e & 63;
        As[r * 65 + c] = x[(size_t)(row0 + r) * CX_ + c];
    }
    __syncthreads();

    v8f acc = {};
#pragma unroll
    for (int kk = 0; kk < 16; ++kk) {
        v2f a, bb;
        a.x  = As[lm * 65 + kk * 4 + 2 * half + 0];
        a.y  = As[lm * 65 + kk * 4 + 2 * half + 1];
        bb.x = W[(size_t)(kk * 4 + 2 * half + 0) * COUT_ + col0 + lm];
        bb.y = W[(size_t)(kk * 4 + 2 * half + 1) * COUT_ + col0 + lm];
        acc = __builtin_amdgcn_wmma_f32_16x16x4_f32(
            /*neg_a=*/false, a, /*neg_b=*/false, bb,
            /*c_mod=*/(short)0, acc, /*reuse_a=*/false, /*reuse_b=*/false);
    }

    const float bv = bias[col0 + lm];
    float s = 0.0f, ss = 0.0f;
#pragma unroll
    for (int r = 0; r < 8; ++r) {
        float hv = acc[r] + bv;
        int row = row0 + half * 8 + r;
        h[(size_t)row * COUT_ + col0 + lm] = hv;
        s += hv; ss += hv * hv;
    }
    // fold lane pairs sharing a column (l and l+16), then one atomic per column
    s  += __shfl_xor(s, 16, 32);
    ss += __shfl_xor(ss, 16, 32);
    if (lane < 16) {
        atomicAdd(&sum[col0 + lane], s);
        atomicAdd(&sumsq[col0 + lane], ss);
    }
}

// ---------------------------------------------------------------------------
// BN batch stats -> per-channel scale/shift (train mode, population var).
// ---------------------------------------------------------------------------
__global__ void finalize_stats_kernel(
    const float* __restrict__ sum, const float* __restrict__ sumsq,
    const float* __restrict__ gamma, const float* __restrict__ beta,
    float* __restrict__ scale, float* __restrict__ shift)
{
    int c = threadIdx.x;
    if (c < COUT_) {
        float mu  = sum[c]   * (1.0f / (float)P_);
        float var = sumsq[c] * (1.0f / (float)P_) - mu * mu;
        float inv = rsqrtf(var + EPS_);
        float sc  = gamma[c] * inv;
        scale[c] = sc;
        shift[c] = beta[c] - mu * sc;
    }
}

// ---------------------------------------------------------------------------
// kNN: block = 256 centers of one cloud; pos streamed through LDS in
// 1024-point tiles with DOUBLE-BUFFERED ASYNC global->LDS copies
// (global_load_async_to_lds_b128, ASYNCcnt-tracked) so the next tile is in
// flight while the wave does the distance/top-16 VALU work.
// Register-resident sorted top-16 (fully unrolled, threshold-guarded).
// ---------------------------------------------------------------------------
#define TILE_   1024
#define NTILE_  (N_ / TILE_)        // 4
#define TBYTES_ (TILE_ * 3 * 4)     // 12288 bytes = 768 x B128 = 256 thr x 3

static __device__ __forceinline__ void issue_tile_async(
    unsigned ldsbase, unsigned long long gbase, int t, int tid)
{
#pragma unroll
    for (int i = 0; i < 3; ++i) {
        unsigned loff = ldsbase + (unsigned)(tid + i * 256) * 16u;
        unsigned long long ga = gbase
            + (unsigned long long)t * (unsigned long long)TBYTES_
            + (unsigned long long)(tid + i * 256) * 16ull;
        asm volatile("global_load_async_to_lds_b128 %0, %1, off"
                     :
                     : "v"(loff), "v"(ga)
                     : "memory");
    }
}

__global__ __launch_bounds__(256) void knn_kernel(
    const float* __restrict__ pos, const float* __restrict__ centers,
    int* __restrict__ knn)
{
    __shared__ float tile[2][TILE_ * 3];   // 2 x 12 KB double buffer
    const int cloud = blockIdx.x >> 2;     // M_/256 == 4 segments per cloud
    const int seg   = blockIdx.x & 3;
    const int tid   = threadIdx.x;
    const int base  = cloud * N_;
    const int cl    = cloud * M_ + seg * 256 + tid;

    const float cx = centers[(size_t)cl * 3 + 0];
    const float cy = centers[(size_t)cl * 3 + 1];
    const float cz = centers[(size_t)cl * 3 + 2];

    const unsigned lds0 = lds_off_u32(&tile[0][0]);
    const unsigned lds1 = lds_off_u32(&tile[1][0]);
    const unsigned long long gbase =
        (unsigned long long)(const void*)(pos + (size_t)base * 3);

    float dk[K_]; int ik[K_];
#pragma unroll
    for (int q = 0; q < K_; ++q) { dk[q] = 3.4e38f; ik[q] = 0; }

    issue_tile_async(lds0, gbase, 0, tid);          // prime buffer 0

    for (int t = 0; t < NTILE_; ++t) {
        asm volatile("s_wait_asynccnt 0x0" ::: "memory");
        __syncthreads();                            // tile t resident for all
        if (t + 1 < NTILE_)                          // overlap: next tile in flight
            issue_tile_async(((t + 1) & 1) ? lds1 : lds0, gbase, t + 1, tid);

        const float* tb = tile[t & 1];
        for (int j = 0; j < TILE_; ++j) {
            float dx = tb[j * 3 + 0] - cx;
            float dy = tb[j * 3 + 1] - cy;
            float dz = tb[j * 3 + 2] - cz;
            float d2 = dx * dx + dy * dy + dz * dz;
            if (d2 < dk[K_ - 1]) {                   // rare insertion path
                float dq = d2; int iq = t * TILE_ + j;
#pragma unroll
                for (int q = 0; q < K_; ++q) {
                    if (dq < dk[q]) {
                        float td = dk[q]; int ti = ik[q];
                        dk[q] = dq; ik[q] = iq;
                        dq = td; iq = ti;
                    }
                }
            }
        }
        __syncthreads();                            // reads done before overwrite
    }
#pragma unroll
    for (int q = 0; q < K_; ++q) knn[(size_t)cl * K_ + q] = base + ik[q];
}

// ---------------------------------------------------------------------------
// Gather + pool: one block per cluster. x_out = max_k relu(scale*h+shift),
// aux_out = mean_k aux. Neighbor list staged in LDS.
// ---------------------------------------------------------------------------
__global__ __launch_bounds__(128) void pool_kernel(
    const float* __restrict__ h, const float* __restrict__ aux,
    const int* __restrict__ knn, const float* __restrict__ scale,
    const float* __restrict__ shift, float* __restrict__ x_out,
    float* __restrict__ aux_out)
{
    __shared__ int nb[K_];
    const int cl  = blockIdx.x;
    const int tid = threadIdx.x;
    if (tid < K_) nb[tid] = knn[(size_t)cl * K_ + tid];
    __syncthreads();

    const float sc = scale[tid], sh = shift[tid];
    float xm = -3.4e38f;
#pragma unroll
    for (int k = 0; k < K_; ++k) {
        float hv = h[(size_t)nb[k] * COUT_ + tid];
        xm = fmaxf(xm, fmaxf(sc * hv + sh, 0.0f));
    }
    x_out[(size_t)cl * COUT_ + tid] = xm;

    if (tid < CAUX_) {
        float am = 0.0f;
#pragma unroll
        for (int k = 0; k < K_; ++k) am += aux[(size_t)nb[k] * CAUX_ + tid];
        aux_out[(size_t)cl * CAUX_ + tid] = am * (1.0f / (float)K_);
    }
}

// ---------------------------------------------------------------------------
// Launch
// ---------------------------------------------------------------------------
extern "C" void kernel_launch(void* const* d_in, const int* in_sizes, int n_in,
                              void* d_out, int out_size, void* d_ws, size_t ws_size,
                              hipStream_t stream) {
    const float* x     = (const float*)d_in[0];
    const float* pos   = (const float*)d_in[1];
    // d_in[2] = batch (int32): clouds are equal-size; geometry implied by B_/N_.
    const float* aux   = (const float*)d_in[3];
    const float* W     = (const float*)d_in[4];
    const float* bias  = (const float*)d_in[5];
    const float* gamma = (const float*)d_in[6];
    const float* beta  = (const float*)d_in[7];

    // workspace layout (floats unless noted):
    float* h       = (float*)d_ws;                        // P*COUT   = 8,388,608
    float* centers = h + (size_t)P_ * COUT_;              // BM*3     =    49,152
    int*   knn     = (int*)(centers + (size_t)BM_ * 3);   // BM*K ints=   262,144
    float* sum     = (float*)(knn + (size_t)BM_ * K_);    // COUT
    float* sumsq   = sum + COUT_;                          // COUT
    float* scale   = sumsq + COUT_;                        // COUT
    float* shift   = scale + COUT_;                        // COUT

    // output layout (flat, return order): x_out, sub_pos, sub_batch, aux_out, id
    float* out       = (float*)d_out;
    float* x_out     = out;                               // BM*COUT
    float* sub_pos   = x_out + (size_t)BM_ * COUT_;       // BM*3
    float* sub_batch = sub_pos + (size_t)BM_ * 3;         // BM
    float* aux_out   = sub_batch + BM_;                   // BM*CAUX
    float* id_out    = aux_out + (size_t)BM_ * CAUX_;     // BM

    hipLaunchKernelGGL(init_stats_kernel, dim3(1), dim3(256), 0, stream, sum, sumsq);
    hipLaunchKernelGGL(fps_kernel, dim3(B_), dim3(1024), 0, stream,
                       pos, centers, sub_pos, sub_batch, id_out);
    hipLaunchKernelGGL(gemm_bn_kernel, dim3(P_ / 16), dim3(256), 0, stream,
                       x, W, bias, h, sum, sumsq);
    hipLaunchKernelGGL(finalize_stats_kernel, dim3(1), dim3(128), 0, stream,
                       sum, sumsq, gamma, beta, scale, shift);
    hipLaunchKernelGGL(knn_kernel, dim3(B_ * (M_ / 256)), dim3(256), 0, stream,
                       pos, centers, knn);
    hipLaunchKernelGGL(pool_kernel, dim3(BM_), dim3(128), 0, stream,
                       h, aux, knn, scale, shift, x_out, aux_out);
}